// GCN_layer_50027779064032
// MI455X (gfx1250) — compile-verified
//
#include <hip/hip_runtime.h>

#define N_NODES 100000
#define N_EDGES 1600000
#define FEATS 128
#define BN_EPS 1e-5f

typedef __attribute__((ext_vector_type(2))) float v2f;
typedef __attribute__((ext_vector_type(8))) float v8f;

// ---------------------------------------------------------------------------
// Kernel 0: zero the 256 stat accumulators (sums[128] + sumsq[128] contiguous)
// ---------------------------------------------------------------------------
__global__ void zero_stats_kernel(float* __restrict__ stats) {
    stats[threadIdx.x] = 0.0f;
}

// ---------------------------------------------------------------------------
// Kernel 1: h = X @ W^T + b  via V_WMMA_F32_16X16X4_F32 (fp32 matrix pipe).
// Each wave computes a 16(node-rows) x 128(out-feats) tile:
//   - 8 x v8f accumulators initialized with bias
//   - A frag (16x4 f32): lane L<16 -> row=L, K=k0..k0+1 ; lane>=16 -> K=k0+2..3
//   - B frag (4x16 f32) = W^T read from LDS: B[k][n] = W[n][k]
// Also writes agg = h (self-loop initialization of the aggregation buffer).
// ---------------------------------------------------------------------------
__global__ __launch_bounds__(256) void gemm_kernel(
    const float* __restrict__ X, const float* __restrict__ W,
    const float* __restrict__ bias, float* __restrict__ h,
    float* __restrict__ agg) {
    __shared__ float Wlds[FEATS * FEATS];  // 64 KB, Wlds[o*128+k] = W[o][k]

    const int tid = threadIdx.x;
    // Cooperative load of W into LDS: 4096 float4, 16 per thread
    {
        const float4* Wv = (const float4*)W;
        float4* Lv = (float4*)Wlds;
        for (int i = tid; i < (FEATS * FEATS) / 4; i += 256) Lv[i] = Wv[i];
    }
    __syncthreads();

    const int wave = tid >> 5;   // 0..7
    const int lane = tid & 31;
    const int l15  = lane & 15;
    const int hi   = lane >> 4;  // 0 or 1 (selects K pair within fragment)
    const int rowbase = (blockIdx.x * 8 + wave) * 16;
    if (rowbase >= N_NODES) return;  // no barriers after this point

    // Accumulators start at the bias (column-dependent only)
    v8f acc[8];
#pragma unroll
    for (int nt = 0; nt < 8; ++nt) {
        const float bv = bias[nt * 16 + l15];
        v8f c;
#pragma unroll
        for (int i = 0; i < 8; ++i) c[i] = bv;
        acc[nt] = c;
    }

    const float* arow = X + (size_t)(rowbase + l15) * FEATS + hi * 2;
#pragma unroll 4
    for (int k0 = 0; k0 < FEATS; k0 += 4) {
        const v2f a = *(const v2f*)(arow + k0);
#pragma unroll
        for (int nt = 0; nt < 8; ++nt) {
            const v2f bf = *(const v2f*)&Wlds[(nt * 16 + l15) * FEATS + k0 + hi * 2];
            acc[nt] = __builtin_amdgcn_wmma_f32_16x16x4_f32(
                false, a, false, bf, (short)0, acc[nt], false, false);
        }
    }

    // C/D layout: lane L<16 -> VGPR i = D[M=i][N=L]; lane>=16 -> D[M=i+8][N=L-16]
    const int r0 = rowbase + hi * 8;
#pragma unroll
    for (int nt = 0; nt < 8; ++nt) {
        const int col = nt * 16 + l15;
#pragma unroll
        for (int i = 0; i < 8; ++i) {
            const float v = acc[nt][i];
            const size_t idx = (size_t)(r0 + i) * FEATS + col;
            h[idx] = v;
            agg[idx] = v;   // self-loop: agg starts at h
        }
    }
}

// ---------------------------------------------------------------------------
// Kernel 2: edge aggregation. One wave per edge iteration: lane l carries
// 4 contiguous floats (512 B coalesced gather per wave from L2-resident h),
// then 4 global_atomic_add_f32 into agg[dst]. Both buffers fit in 192 MB L2.
// ---------------------------------------------------------------------------
__global__ __launch_bounds__(256) void edge_kernel(
    const int* __restrict__ src, const int* __restrict__ dst,
    const float* __restrict__ h, float* __restrict__ agg) {
    const int lane = threadIdx.x & 31;
    const int wave = (int)((blockIdx.x * blockDim.x + threadIdx.x) >> 5);
    const int nwaves = (int)((gridDim.x * blockDim.x) >> 5);
    for (int e = wave; e < N_EDGES; e += nwaves) {
        const int s = src[e];
        const int d = dst[e];
        const float4 v = *(const float4*)(h + (size_t)s * FEATS + lane * 4);
        float* ap = agg + (size_t)d * FEATS + lane * 4;
        atomicAdd(ap + 0, v.x);
        atomicAdd(ap + 1, v.y);
        atomicAdd(ap + 2, v.z);
        atomicAdd(ap + 3, v.w);
    }
}

// ---------------------------------------------------------------------------
// Kernel 3: per-column sum / sum-of-squares. Thread owns col = tid&127,
// block strides 2 rows/iter (fully coalesced 512 B row reads), LDS fold,
// then 256 global atomics per block into the 128-entry accumulators.
// ---------------------------------------------------------------------------
__global__ __launch_bounds__(256) void stats_kernel(
    const float* __restrict__ agg, float* __restrict__ sums,
    float* __restrict__ sumsq) {
    __shared__ float s1[256];
    __shared__ float s2[256];
    const int tid = threadIdx.x;
    const int col = tid & 127;
    const int half = tid >> 7;
    float a = 0.0f, b = 0.0f;
    for (int row = blockIdx.x * 2 + half; row < N_NODES; row += gridDim.x * 2) {
        const float v = agg[(size_t)row * FEATS + col];
        a += v;
        b += v * v;
    }
    s1[tid] = a;
    s2[tid] = b;
    __syncthreads();
    if (tid < 128) {
        atomicAdd(&sums[col], s1[tid] + s1[tid + 128]);
        atomicAdd(&sumsq[col], s2[tid] + s2[tid + 128]);
    }
}

// ---------------------------------------------------------------------------
// Kernel 4: batchnorm (population stats) + ReLU + residual, in-place on out.
// ---------------------------------------------------------------------------
__global__ __launch_bounds__(256) void finalize_kernel(
    float* __restrict__ out, const float* __restrict__ x,
    const float* __restrict__ gamma, const float* __restrict__ beta,
    const float* __restrict__ sums, const float* __restrict__ sumsq) {
    const size_t total = (size_t)N_NODES * FEATS;
    const size_t stride = (size_t)gridDim.x * blockDim.x;
    for (size_t i = (size_t)blockIdx.x * blockDim.x + threadIdx.x; i < total;
         i += stride) {
        const int col = (int)(i & (FEATS - 1));
        const float mean = sums[col] * (1.0f / N_NODES);
        const float var = sumsq[col] * (1.0f / N_NODES) - mean * mean;
        const float rs = rsqrtf(var + BN_EPS);
        float v = gamma[col] * (out[i] - mean) * rs + beta[col];
        v = fmaxf(v, 0.0f);
        out[i] = v + x[i];
    }
}

// ---------------------------------------------------------------------------
extern "C" void kernel_launch(void* const* d_in, const int* in_sizes, int n_in,
                              void* d_out, int out_size, void* d_ws,
                              size_t ws_size, hipStream_t stream) {
    const float* node_feats = (const float*)d_in[0];
    const float* W          = (const float*)d_in[1];
    const float* b          = (const float*)d_in[2];
    const float* gamma      = (const float*)d_in[3];
    const float* beta       = (const float*)d_in[4];
    const int*   src        = (const int*)d_in[5];
    const int*   dst        = (const int*)d_in[6];
    float* out = (float*)d_out;

    // Workspace: h [N_NODES*128 f32] | sums[128] | sumsq[128]
    float* h     = (float*)d_ws;
    float* sums  = h + (size_t)N_NODES * FEATS;
    float* sumsq = sums + FEATS;

    zero_stats_kernel<<<1, 256, 0, stream>>>(sums);

    // 6250 row-tiles of 16, 8 waves (tiles) per 256-thread block
    const int gemm_blocks = (N_NODES / 16 + 7) / 8;  // 782
    gemm_kernel<<<gemm_blocks, 256, 0, stream>>>(node_feats, W, b, h, out);

    edge_kernel<<<4096, 256, 0, stream>>>(src, dst, h, out);

    stats_kernel<<<1024, 256, 0, stream>>>(out, sums, sumsq);

    finalize_kernel<<<8192, 256, 0, stream>>>(out, node_feats, gamma, beta,
                                              sums, sumsq);
}